// GraphAttentionLayer_34351148433905
// MI455X (gfx1250) — compile-verified
//
#include <hip/hip_runtime.h>
#include <hip/hip_bf16.h>
#include <stdint.h>

// Problem constants (match reference)
#define GAT_B      2
#define GAT_N      20000
#define GAT_E      200000
#define GAT_INF    256
#define GAT_OUTF   32
#define GAT_HEADS  4
#define GAT_OUTC   (GAT_OUTF * GAT_HEADS)   // 128
#define NEG_SLOPE  0.2f

#define TILE_M     128
#define LDS_STRIDE 260    // 256 + 4-dword pad (TDM pad_enable) -> conflict-free ds_load_b64

typedef __attribute__((ext_vector_type(2))) float v2f;
typedef __attribute__((ext_vector_type(8))) float v8f;

#if defined(__gfx1250__)
typedef __attribute__((ext_vector_type(4))) unsigned int v4u;
typedef __attribute__((ext_vector_type(8))) int          v8i;
typedef __attribute__((ext_vector_type(4))) int          v4i;

// TDM: load a (rows x 256) f32 tile, row stride 256 in global, into LDS with
// +4 dwords of padding after every 256 dwords (-> LDS row stride 260 floats).
// Rows >= rows_valid are zero-filled by TDM OOB handling.
__device__ __forceinline__ void tdm_load_tile_256xR(const float* gsrc,
                                                    unsigned lds_byte_off,
                                                    unsigned rows_valid)
{
  unsigned long long ga = (unsigned long long)(uintptr_t)gsrc;
  v4u g0;
  g0.x = 1u;                                         // count = 1 (valid D#)
  g0.y = lds_byte_off;                               // lds_addr
  g0.z = (unsigned)(ga & 0xFFFFFFFFull);             // global_addr[31:0]
  g0.w = (unsigned)((ga >> 32) & 0x01FFFFFFull)      // global_addr[56:32]
       | (2u << 30);                                 // type = 2 ("image")
  v8i g1;
  g1.s0 = (int)((2u << 16)      // data_size = 4B
              | (1u << 20)      // pad_enable
              | (7u << 22)      // pad_interval: every 256 dwords
              | (3u << 25));    // pad_amount: 4 dwords
  g1.s1 = (int)(256u << 16);                                   // tensor_dim0 = 256
  g1.s2 = (int)((rows_valid & 0xFFFFu) << 16);                 // tensor_dim1 lo16
  g1.s3 = (int)(((rows_valid >> 16) & 0xFFFFu) | (256u << 16));// dim1 hi16 | tile_dim0=256
  g1.s4 = (int)128u;                                           // tile_dim1 = 128, tile_dim2 = 0
  g1.s5 = 256;                                                 // tensor_dim0_stride = 256
  g1.s6 = 0;
  g1.s7 = 0;
  v4i z4 = {0, 0, 0, 0};
#if __clang_major__ >= 23
  v8i z8 = {0, 0, 0, 0, 0, 0, 0, 0};
  __builtin_amdgcn_tensor_load_to_lds(g0, g1, z4, z4, z8, 0);
#else
  __builtin_amdgcn_tensor_load_to_lds(g0, g1, z4, z4, 0);
#endif
}
#endif // __gfx1250__

// ---------------------------------------------------------------------------
// Kernel 1: h = x @ W.T   (40000x256) x (256x128) via V_WMMA_F32_16X16X4_F32.
// Block = 256 threads (8 waves). Each wave computes a 16x128 C tile.
// W (128x256) and the 128-row x tile are TDM-staged into padded LDS.
// ---------------------------------------------------------------------------
__global__ __launch_bounds__(256)
void gat_gemm_wmma(const float* __restrict__ x, const float* __restrict__ W,
                   float* __restrict__ h, int Mtot)
{
#if defined(__gfx1250__)
  extern __shared__ float smem[];
  float* Ws = smem;                          // [128][LDS_STRIDE]  (out-col major)
  float* Xs = smem + GAT_OUTC * LDS_STRIDE;  // [TILE_M][LDS_STRIDE]
  const int m0 = blockIdx.x * TILE_M;

  if (threadIdx.x == 0) {
    tdm_load_tile_256xR(W, (unsigned)(uintptr_t)Ws, (unsigned)GAT_OUTC);
    tdm_load_tile_256xR(x + (size_t)m0 * GAT_INF, (unsigned)(uintptr_t)Xs,
                        (unsigned)(Mtot - m0));
    __builtin_amdgcn_s_wait_tensorcnt(0);
  }
  __syncthreads();

  const int warp = threadIdx.x >> 5;
  const int lane = threadIdx.x & 31;
  const int half = lane >> 4;    // 0: K0/K1 half, 1: K2/K3 half (A/B frags)
  const int l15  = lane & 15;

  // A frag source: row (warp*16 + l15) of Xs; pairs (k, k+1) at float2 index half + k/2
  const float2* arow =
      reinterpret_cast<const float2*>(Xs + (warp * 16 + l15) * LDS_STRIDE) + half;
  // B frag sources: out-col (j*16 + l15) rows of Ws
  const float2* brow[8];
#pragma unroll
  for (int j = 0; j < 8; ++j)
    brow[j] = reinterpret_cast<const float2*>(Ws + (j * 16 + l15) * LDS_STRIDE) + half;

  v8f c[8];
#pragma unroll
  for (int j = 0; j < 8; ++j)
    c[j] = (v8f){0.f, 0.f, 0.f, 0.f, 0.f, 0.f, 0.f, 0.f};

  // K loop: 4 f32 K-values per WMMA -> 64 iterations
  for (int k2 = 0; k2 < GAT_INF / 2; k2 += 2) {
    float2 av = arow[k2];
    v2f a; a.x = av.x; a.y = av.y;
#pragma unroll
    for (int j = 0; j < 8; ++j) {
      float2 bv = brow[j][k2];
      v2f b; b.x = bv.x; b.y = bv.y;
      c[j] = __builtin_amdgcn_wmma_f32_16x16x4_f32(
          false, a, false, b, (short)0, c[j], false, false);
    }
  }

  // C/D layout: vgpr v, lanes 0-15 -> M=v, N=lane; lanes 16-31 -> M=8+v
  const int mb = m0 + warp * 16 + 8 * half;
#pragma unroll
  for (int j = 0; j < 8; ++j) {
#pragma unroll
    for (int v = 0; v < 8; ++v) {
      const int mg = mb + v;
      if (mg < Mtot)
        h[(size_t)mg * GAT_OUTC + j * 16 + l15] = c[j][v];
    }
  }
#endif
}

// ---------------------------------------------------------------------------
// Kernel 2: init out = 0, last_edge = -1 (d_ws/d_out arrive poisoned)
// ---------------------------------------------------------------------------
__global__ void gat_init(float* __restrict__ out, int* __restrict__ last_edge,
                         int outN, int N)
{
  const int t = blockIdx.x * blockDim.x + threadIdx.x;
  if (t < outN) out[t] = 0.f;
  if (t < N) last_edge[t] = -1;
}

// ---------------------------------------------------------------------------
// Kernel 3: last_edge[n] = segment_max(edge_id, src)
// ---------------------------------------------------------------------------
__global__ void gat_edge_max(const int* __restrict__ edges,
                             int* __restrict__ last_edge, int E)
{
  const int e = blockIdx.x * blockDim.x + threadIdx.x;
  if (e < E) atomicMax(&last_edge[edges[e]], e);
}

// ---------------------------------------------------------------------------
// Kernel 4: per-node attention weights. Scores only needed at last_edge[n];
// leaky-relu then softmax over the batch axis (2 entries).
// ---------------------------------------------------------------------------
__global__ __launch_bounds__(256)
void gat_node_aw(const float* __restrict__ h, const int* __restrict__ edges,
                 const int* __restrict__ last_edge, const float* __restrict__ a,
                 float* __restrict__ aw, int N, int E)
{
  const int n = blockIdx.x * blockDim.x + threadIdx.x;
  if (n >= N) return;
  const int e = last_edge[n];
  if (e < 0) {
#pragma unroll
    for (int b = 0; b < GAT_B; ++b)
#pragma unroll
      for (int hh = 0; hh < GAT_HEADS; ++hh)
        aw[((size_t)(b * N + n)) * GAT_HEADS + hh] = 0.f;
    return;
  }
  const int d = edges[E + e];  // dst of the last edge (src is n by construction)
  float s[GAT_B][GAT_HEADS];
#pragma unroll
  for (int b = 0; b < GAT_B; ++b) {
    const float* hs = h + ((size_t)(b * N + n)) * GAT_OUTC;
    const float* hd = h + ((size_t)(b * N + d)) * GAT_OUTC;
#pragma unroll
    for (int hh = 0; hh < GAT_HEADS; ++hh) {
      float acc = 0.f;
      for (int f = 0; f < GAT_OUTF; ++f)
        acc += hs[hh * GAT_OUTF + f] * a[f] + hd[hh * GAT_OUTF + f] * a[GAT_OUTF + f];
      s[b][hh] = (acc > 0.f) ? acc : NEG_SLOPE * acc;
    }
  }
#pragma unroll
  for (int hh = 0; hh < GAT_HEADS; ++hh) {
    const float m  = fmaxf(s[0][hh], s[1][hh]);
    const float e0 = __expf(s[0][hh] - m);
    const float e1 = __expf(s[1][hh] - m);
    const float inv = 1.f / (e0 + e1);
    aw[((size_t)(0 * N + n)) * GAT_HEADS + hh] = e0 * inv;
    aw[((size_t)(1 * N + n)) * GAT_HEADS + hh] = e1 * inv;
  }
}

// ---------------------------------------------------------------------------
// Kernel 5: fused weighted scatter + head mean:
//   out[b,src,f] += 0.25 * sum_h aw[b,src,h] * h[b,dst,h*32+f]
// 32 lanes per (edge,batch) pair: lane = f, coalesced h reads, f32 atomics
// into the L2-resident output buffer.
// ---------------------------------------------------------------------------
__global__ __launch_bounds__(256)
void gat_edge_agg(const float* __restrict__ h, const int* __restrict__ edges,
                  const float* __restrict__ aw, float* __restrict__ out,
                  int E, int N)
{
  const long long t = (long long)blockIdx.x * blockDim.x + threadIdx.x;
  const int f = (int)(t & 31);
  const long long p = t >> 5;                 // pair index over E*B
  if (p >= (long long)E * GAT_B) return;
  const int b = (int)(p & 1);
  const int e = (int)(p >> 1);
  const int n = edges[e];                     // src
  const int d = edges[E + e];                 // dst
  const float* hd  = h  + ((size_t)(b * N + d)) * GAT_OUTC;
  const float* awp = aw + ((size_t)(b * N + n)) * GAT_HEADS;
  float acc = 0.f;
#pragma unroll
  for (int hh = 0; hh < GAT_HEADS; ++hh)
    acc += awp[hh] * hd[hh * GAT_OUTF + f];
  atomicAdd(&out[((size_t)(b * N + n)) * GAT_OUTF + f], 0.25f * acc);
}

// ---------------------------------------------------------------------------
extern "C" void kernel_launch(void* const* d_in, const int* in_sizes, int n_in,
                              void* d_out, int out_size, void* d_ws, size_t ws_size,
                              hipStream_t stream)
{
  (void)in_sizes; (void)n_in; (void)out_size; (void)ws_size;

  const float* x     = (const float*)d_in[0];   // (B,N,256) f32
  const int*   edges = (const int*)d_in[1];     // (2,E) int32 (jax x64 off)
  const float* W     = (const float*)d_in[2];   // (128,256) f32
  const float* a     = (const float*)d_in[3];   // (64,1) f32
  float*       out   = (float*)d_out;           // (B,N,32) f32

  const int B = GAT_B, N = GAT_N, E = GAT_E;
  const int M = B * N;                          // 40000 GEMM rows

  // Workspace layout (256B aligned)
  char* ws = (char*)d_ws;
  size_t off = 0;
  float* h = (float*)(ws + off);
  off += ((size_t)M * GAT_OUTC * sizeof(float) + 255) & ~(size_t)255;   // 20.48 MB
  float* aw = (float*)(ws + off);
  off += ((size_t)B * N * GAT_HEADS * sizeof(float) + 255) & ~(size_t)255; // 640 KB
  int* last_edge = (int*)(ws + off);                                     // 80 KB

  const int outN = B * N * GAT_OUTF;            // 1,280,000
  const int initN = (outN > N) ? outN : N;
  gat_init<<<(initN + 255) / 256, 256, 0, stream>>>(out, last_edge, outN, N);

  const size_t shmem = (size_t)(GAT_OUTC + TILE_M) * LDS_STRIDE * sizeof(float); // 266,240 B
  gat_gemm_wmma<<<(M + TILE_M - 1) / TILE_M, 256, shmem, stream>>>(x, W, h, M);

  gat_edge_max<<<(E + 255) / 256, 256, 0, stream>>>(edges, last_edge, E);

  gat_node_aw<<<(N + 255) / 256, 256, 0, stream>>>(h, edges, last_edge, a, aw, N, E);

  const long long aggT = (long long)E * GAT_B * 32;                      // 12.8M threads
  gat_edge_agg<<<(int)((aggT + 255) / 256), 256, 0, stream>>>(h, edges, aw, out, E, N);
}